// ProteinFeaturesDirect_61220463837958
// MI455X (gfx1250) — compile-verified
//
#include <hip/hip_runtime.h>
#include <hip/hip_bf16.h>
#include <math.h>

// ---------------------------------------------------------------------------
// ProteinFeatures on MI455X (gfx1250, wave32).
//  - GEMMs on V_WMMA_F32_16X16X4_F32 (exact fp32 semantics).
//  - 32 edges per block: each wave computes TWO 16x16 C tiles that share one
//    B fragment -> one coalesced global_load_b64 feeds two WMMAs (2x the
//    arithmetic intensity on the L2 path vs one tile per wave).
//  - Weights repacked once into the per-lane WMMA B-operand layout.
//  - RBF (v_exp_f32) + positional features built cooperatively in LDS,
//    LayerNorm fused between the two GEMMs.
// ---------------------------------------------------------------------------

#define LRES   8192
#define KNBR   48
#define EFD    128
#define KFEAT  416       // 16 pos + 25*16 rbf
#define NT1    104       // KFEAT/4 wmma k-steps
#define NT2    32        // EFD/4 wmma k-steps
#define MT     32        // edges per block (two 16-row WMMA tiles)
#define ESTR   418       // padded LDS row stride (floats, even, bank-spread)
#define HSTR   130       // padded LDS row stride for normalized features

typedef __attribute__((ext_vector_type(2))) float v2f;
typedef __attribute__((ext_vector_type(8))) float v8f;

// ---------------------------------------------------------------------------
// Kernel 1: backbone atoms  bb[i][5][3] = {N, Ca, C, O, Cb}
// ---------------------------------------------------------------------------
__global__ __launch_bounds__(256) void bb_kernel(const float* __restrict__ X,
                                                 const float* __restrict__ noise,
                                                 float* __restrict__ bb) {
  int i = blockIdx.x * 256 + threadIdx.x;
  if (i >= LRES) return;
  float a4[4][3];
#pragma unroll
  for (int a = 0; a < 4; ++a)
#pragma unroll
    for (int d = 0; d < 3; ++d) {
      int idx = (i * 4 + a) * 3 + d;
      a4[a][d] = X[idx] + 0.1f * noise[idx];
    }
  float b[3], c[3], cr[3], cb[3];
#pragma unroll
  for (int d = 0; d < 3; ++d) { b[d] = a4[1][d] - a4[0][d]; c[d] = a4[2][d] - a4[1][d]; }
  cr[0] = b[1] * c[2] - b[2] * c[1];
  cr[1] = b[2] * c[0] - b[0] * c[2];
  cr[2] = b[0] * c[1] - b[1] * c[0];
#pragma unroll
  for (int d = 0; d < 3; ++d)
    cb[d] = -0.58273431f * cr[d] + 0.56802827f * b[d] - 0.54067466f * c[d] + a4[1][d];
  float* o = bb + (size_t)i * 15;
#pragma unroll
  for (int d = 0; d < 3; ++d) {
    o[0 * 3 + d] = a4[0][d];
    o[1 * 3 + d] = a4[1][d];
    o[2 * 3 + d] = a4[2][d];
    o[3 * 3 + d] = a4[3][d];
    o[4 * 3 + d] = cb[d];
  }
}

// ---------------------------------------------------------------------------
// Kernel 2: top-K nearest neighbors per residue (distances over Ca).
// One 256-thread block per residue; full distance row in LDS; K argmin passes
// with (dist, index) tie-break matching jax.lax.top_k ordering.
// ---------------------------------------------------------------------------
__global__ __launch_bounds__(256) void topk_kernel(const float* __restrict__ bb,
                                                   const float* __restrict__ mask,
                                                   int* __restrict__ nbrOut) {
  __shared__ float dist[LRES];
  __shared__ float sd[256];
  __shared__ int   si[256];
  const int i = blockIdx.x;
  const int t = threadIdx.x;
  const float cax = bb[(size_t)i * 15 + 3];
  const float cay = bb[(size_t)i * 15 + 4];
  const float caz = bb[(size_t)i * 15 + 5];
  const float mi = mask[i];
  for (int j = t; j < LRES; j += 256) {
    float dx = cax - bb[(size_t)j * 15 + 3];
    float dy = cay - bb[(size_t)j * 15 + 4];
    float dz = caz - bb[(size_t)j * 15 + 5];
    float d = sqrtf(dx * dx + dy * dy + dz * dz + 1e-6f);
    bool pm = (mi * mask[j]) != 0.0f;
    dist[j] = pm ? d : __builtin_inff();
  }
  __syncthreads();
  for (int it = 0; it < KNBR; ++it) {
    float best = __builtin_inff();
    int bidx = 0x7fffffff;
    for (int j = t; j < LRES; j += 256) {
      float d = dist[j];
      if (d < best || (d == best && j < bidx)) { best = d; bidx = j; }
    }
    sd[t] = best; si[t] = bidx;
    __syncthreads();
    for (int s = 128; s > 0; s >>= 1) {
      if (t < s) {
        float dv = sd[t + s]; int iv = si[t + s];
        if (dv < sd[t] || (dv == sd[t] && iv < si[t])) { sd[t] = dv; si[t] = iv; }
      }
      __syncthreads();
    }
    if (t == 0) {
      int w = si[0];
      nbrOut[(size_t)i * KNBR + it] = w;
      dist[w] = __builtin_inff();
    }
    __syncthreads();
  }
}

// ---------------------------------------------------------------------------
// Kernel 3: repack W_e (128x416) and W_proj (128x128) into the per-lane
// v_wmma_f32_16x16x4_f32 B-operand layout:
//   Wp[((t*128 + n)*2 + koff)*2 + j] = W[n][t*4 + 2*koff + j]
// Lane l of wave w reads Wp as v2f at index (t*128 + 16w + (l&15))*2 + (l>>4):
// one contiguous 256B segment per wave per wmma k-step.
// ---------------------------------------------------------------------------
__global__ __launch_bounds__(256) void pack_kernel(const float* __restrict__ We,
                                                   const float* __restrict__ Wproj,
                                                   float* __restrict__ Wp1,
                                                   float* __restrict__ Wp2) {
  int idx = blockIdx.x * 256 + threadIdx.x;
  const int total1 = NT1 * EFD * 4;   // 53248
  const int total2 = NT2 * EFD * 4;   // 16384
  if (idx < total1) {
    int j = idx & 1, koff = (idx >> 1) & 1, n = (idx >> 2) & 127, t = idx >> 9;
    Wp1[idx] = We[(size_t)n * KFEAT + t * 4 + koff * 2 + j];
  } else if (idx < total1 + total2) {
    int r = idx - total1;
    int j = r & 1, koff = (r >> 1) & 1, n = (r >> 2) & 127, t = r >> 9;
    Wp2[r] = Wproj[(size_t)n * EFD + t * 4 + koff * 2 + j];
  }
}

// ---------------------------------------------------------------------------
// Kernel 4: fused edge features + GEMM1 + LayerNorm + GEMM2.
// 32 edges per block, 256 threads = 8 waves. Wave w owns output cols
// [16w, 16w+16) and both edge row-tiles (rows 0-15 and 16-31).
// ---------------------------------------------------------------------------
__global__ __launch_bounds__(256) void edge_kernel(
    const float* __restrict__ bb, const int* __restrict__ nbr,
    const int* __restrict__ resIdx, const int* __restrict__ chainIdx,
    const float* __restrict__ wPosW, const float* __restrict__ wPosB,
    const float* __restrict__ Wp1, const float* __restrict__ nScale,
    const float* __restrict__ nBias, const float* __restrict__ Wp2,
    const float* __restrict__ pBias, float* __restrict__ out) {
  __shared__ float E[MT][ESTR];     // 416 edge features per edge (padded)
  __shared__ float Hh[MT][HSTR];    // 128 normalized features (padded)
  __shared__ float XYZ[MT][32];     // 15 floats of bb[i], then 15 of bb[j]
  __shared__ float sMean[MT], sRstd[MT];

  const int tid = threadIdx.x;
  const int m = tid >> 3;           // edge slot 0..31
  const int s = tid & 7;            // worker within edge
  const int tile = blockIdx.x;
  const int e = tile * MT + m;      // global edge id (< 393216)
  const int i = e / KNBR;
  const int kk = e - i * KNBR;
  const int j = nbr[(size_t)i * KNBR + kk];

  // ---- stage coords (30 floats per edge, 4 per worker) ----
#pragma unroll
  for (int q = 0; q < 4; ++q) {
    int idx = s + 8 * q;
    if (idx < 15)       XYZ[m][idx] = bb[(size_t)i * 15 + idx];
    else if (idx < 30)  XYZ[m][idx] = bb[(size_t)j * 15 + (idx - 15)];
  }
  // ---- positional features (one-hot gather): worker s does feats s, s+8 ----
  {
    int off = resIdx[i] - resIdx[j];
    int same = (chainIdx[i] == chainIdx[j]) ? 1 : 0;
    int offf = off + 32;
    offf = offf < 0 ? 0 : (offf > 64 ? 64 : offf);
    int enc = same ? offf : 65;
#pragma unroll
    for (int q = 0; q < 2; ++q) {
      int d = s + 8 * q;
      E[m][d] = wPosW[d * 66 + enc] + wPosB[d];
    }
  }
  __syncthreads();

  // ---- RBF features: worker s does pairs s, s+8, s+16, s+24 (<25) ----
  for (int p = s; p < 25; p += 8) {
    int a = p / 5, b = p - a * 5;
    float dx = XYZ[m][a * 3 + 0] - XYZ[m][15 + b * 3 + 0];
    float dy = XYZ[m][a * 3 + 1] - XYZ[m][15 + b * 3 + 1];
    float dz = XYZ[m][a * 3 + 2] - XYZ[m][15 + b * 3 + 2];
    float D = sqrtf(dx * dx + dy * dy + dz * dz + 1e-6f);
#pragma unroll
    for (int bin = 0; bin < 16; ++bin) {
      float mu = 2.0f + (float)bin * (20.0f / 15.0f);
      float z = (D - mu) * (1.0f / 1.25f);
      E[m][16 + p * 16 + bin] = __expf(-z * z);
    }
  }
  __syncthreads();

  // ---- WMMA operand addressing ----
  const int wave = tid >> 5;        // 0..7 -> output cols 16*wave..+15
  const int lane = tid & 31;
  const int nloc = lane & 15;
  const int koff = lane >> 4;       // selects K pair {2*koff, 2*koff+1}
  const int n0 = wave * 16;
  const int mrow = nloc;            // A-operand row within each 16-edge tile
  const v2f* __restrict__ Wp1v = (const v2f*)Wp1;
  const v2f* __restrict__ Wp2v = (const v2f*)Wp2;

  // ---- GEMM1: two [16 x 416]x[416 x 16] tiles per wave, shared B ----
  v8f acc0 = {0.f, 0.f, 0.f, 0.f, 0.f, 0.f, 0.f, 0.f};
  v8f acc1 = {0.f, 0.f, 0.f, 0.f, 0.f, 0.f, 0.f, 0.f};
#pragma unroll 4
  for (int t = 0; t < NT1; ++t) {
    int kb = t * 4 + koff * 2;
    v2f bbv = Wp1v[(size_t)(t * EFD + n0 + nloc) * 2 + koff];
    v2f a0, a1;
    a0.x = E[mrow][kb];       a0.y = E[mrow][kb + 1];
    a1.x = E[16 + mrow][kb];  a1.y = E[16 + mrow][kb + 1];
    acc0 = __builtin_amdgcn_wmma_f32_16x16x4_f32(false, a0, false, bbv,
                                                 (short)0, acc0, false, false);
    acc1 = __builtin_amdgcn_wmma_f32_16x16x4_f32(false, a1, false, bbv,
                                                 (short)0, acc1, false, false);
  }
  // C/D layout: VGPR v -> row v (lanes 0-15) / row v+8 (lanes 16-31), col = lane&15
  {
    int n = n0 + nloc;
    int mbase = koff * 8;
#pragma unroll
    for (int v = 0; v < 8; ++v) {
      Hh[mbase + v][n]      = acc0[v];
      Hh[16 + mbase + v][n] = acc1[v];
    }
  }
  __syncthreads();

  // ---- LayerNorm over the 128 features of each edge ----
  if (tid < MT) {
    float sum = 0.f, sumsq = 0.f;
    for (int f = 0; f < EFD; ++f) {
      float x = Hh[tid][f];
      sum += x; sumsq += x * x;
    }
    float mean = sum * (1.0f / 128.0f);
    float var = sumsq * (1.0f / 128.0f) - mean * mean;
    sMean[tid] = mean;
    sRstd[tid] = rsqrtf(var + 1e-5f);
  }
  __syncthreads();
  {
    float mean = sMean[m], rstd = sRstd[m];
#pragma unroll
    for (int q = 0; q < 16; ++q) {
      int f = s + 8 * q;
      Hh[m][f] = (Hh[m][f] - mean) * rstd * nScale[f] + nBias[f];
    }
  }
  __syncthreads();

  // ---- GEMM2: two [16 x 128]x[128 x 16] tiles per wave, shared B,
  //      accumulators seeded with the projection bias ----
  v8f acc2, acc3;
  {
    float bi = pBias[n0 + nloc];
#pragma unroll
    for (int v = 0; v < 8; ++v) { acc2[v] = bi; acc3[v] = bi; }
  }
#pragma unroll 4
  for (int t = 0; t < NT2; ++t) {
    int kb = t * 4 + koff * 2;
    v2f bbv = Wp2v[(size_t)(t * EFD + n0 + nloc) * 2 + koff];
    v2f a0, a1;
    a0.x = Hh[mrow][kb];       a0.y = Hh[mrow][kb + 1];
    a1.x = Hh[16 + mrow][kb];  a1.y = Hh[16 + mrow][kb + 1];
    acc2 = __builtin_amdgcn_wmma_f32_16x16x4_f32(false, a0, false, bbv,
                                                 (short)0, acc2, false, false);
    acc3 = __builtin_amdgcn_wmma_f32_16x16x4_f32(false, a1, false, bbv,
                                                 (short)0, acc3, false, false);
  }
  // ---- store: out[edge][n] ----
  {
    int n = n0 + nloc;
    int mbase = koff * 8;
    size_t base = (size_t)tile * MT + mbase;
#pragma unroll
    for (int v = 0; v < 8; ++v) {
      out[(base + v) * EFD + n]      = acc2[v];
      out[(base + 16 + v) * EFD + n] = acc3[v];
    }
  }
}

// ---------------------------------------------------------------------------
extern "C" void kernel_launch(void* const* d_in, const int* in_sizes, int n_in,
                              void* d_out, int out_size, void* d_ws, size_t ws_size,
                              hipStream_t stream) {
  const float* coords   = (const float*)d_in[0];
  const float* noise    = (const float*)d_in[1];
  const float* mask     = (const float*)d_in[2];
  const int*   resIdx   = (const int*)d_in[3];
  const int*   chainIdx = (const int*)d_in[4];
  const float* wPosW    = (const float*)d_in[5];
  const float* wPosB    = (const float*)d_in[6];
  const float* We       = (const float*)d_in[7];
  const float* nScale   = (const float*)d_in[8];
  const float* nBias    = (const float*)d_in[9];
  const float* Wproj    = (const float*)d_in[10];
  const float* pBias    = (const float*)d_in[11];

  float* efOut = (float*)d_out;
  int*   nbrOut = (int*)(efOut + (size_t)LRES * KNBR * EFD);

  char* ws = (char*)d_ws;
  float* bb  = (float*)ws;                                    // 8192*15*4   = 491520 B
  float* Wp1 = (float*)(ws + 491520);                         // 104*128*4*4 = 212992 B
  float* Wp2 = (float*)(ws + 491520 + 212992);                // 32*128*4*4  =  65536 B

  bb_kernel<<<LRES / 256, 256, 0, stream>>>(coords, noise, bb);
  pack_kernel<<<(NT1 * EFD * 4 + NT2 * EFD * 4) / 256, 256, 0, stream>>>(We, Wproj, Wp1, Wp2);
  topk_kernel<<<LRES, 256, 0, stream>>>(bb, mask, nbrOut);
  edge_kernel<<<(LRES * KNBR) / MT, 256, 0, stream>>>(
      bb, nbrOut, resIdx, chainIdx, wPosW, wPosB, Wp1, nScale, nBias, Wp2, pBias, efOut);
}